// UpsampleLayer_12472585027907
// MI455X (gfx1250) — compile-verified
//
#include <hip/hip_runtime.h>

typedef __attribute__((ext_vector_type(2))) float v2f;
typedef __attribute__((ext_vector_type(8))) float v8f;

#define CIN   512
#define COUT  512
#define HH    64
#define WW    64
#define NPIX  (HH * WW)       // 4096
#define H2    128
#define W2    128

#define BM 64
#define BN 128
#define BK 16
#define WT_STRIDE 20          // padded W-tile row stride (floats): conflict-free b64 A reads
#define XT_STRIDE 288         // K-pair-interleaved X-tile row stride: 2*BN + 32, == 32 mod 64
#define KPAIRS (BK / 2)       // 8 k-pair rows per chunk

__device__ __forceinline__ float sanitize(float v) {
    // nan->0, +inf->1, -inf->-1 ; branchless -> v_cmp + v_cndmask
    float r = (v == v) ? v : 0.0f;
    r = (r ==  __builtin_inff()) ?  1.0f : r;
    r = (r == -__builtin_inff()) ? -1.0f : r;
    return r;
}

// -------------------------------------------------------------------------
// Kernel 1: channel-mix GEMM at low resolution using V_WMMA_F32_16X16X4_F32
//   Z[b, o, p] = sum_c W[o, c] * sanitize(x[b, c, p])
// Grid: (NPIX/BN, COUT/BM, B), Block: 256 threads (8 wave32 waves).
// Each wave computes a 16x64 strip (4 accumulators): 1 A-frag feeds 4 WMMAs.
// X tile stored K-pair interleaved so each B fragment is a single ds_load_b64.
// Ping-pong LDS, one barrier per K-chunk.
// -------------------------------------------------------------------------
__global__ __launch_bounds__(256) void gemm_cmix_kernel(
    const float* __restrict__ x,     // [B, CIN, NPIX]
    const float* __restrict__ wmat,  // [COUT, CIN]
    float* __restrict__ z)           // [B, COUT, NPIX]
{
    __shared__ float Wt[2][BM * WT_STRIDE];        // 2 x 5120 B
    __shared__ float Xt[2][KPAIRS * XT_STRIDE];    // 2 x 9216 B

    const int tid  = threadIdx.x;
    const int lane = tid & 31;
    const int wave = tid >> 5;
    const int half = lane >> 4;          // 0: lanes 0-15, 1: lanes 16-31
    const int l16  = lane & 15;

    const int b  = blockIdx.z;
    const int m0 = blockIdx.y * BM;      // Cout tile base
    const int n0 = blockIdx.x * BN;      // pixel tile base

    const int wm = (wave & 3) * 16;      // wave M offset within block tile
    const int wn = (wave >> 2) * 64;     // wave N offset (four 16x16 tiles)

    v8f c0 = {}, c1 = {}, c2 = {}, c3 = {};

    // ---- global load assignments ----
    // W tile: BM x BK = 1024 floats -> one float4 per thread
    const int lw_row = tid >> 2;             // 0..63
    const int lw_col = (tid & 3) * 4;        // 0..12
    // X tile: BK x BN = 2048 floats -> two float4 per thread (rows r and r+8)
    const int lx_row = tid >> 5;             // 0..7 (wave id: whole wave on one row)
    const int lx_col = (tid & 31) * 4;       // 0..124, fully coalesced 512B/row

    const float* wg = wmat + (size_t)(m0 + lw_row) * CIN + lw_col;
    const float* xg = x + (size_t)b * CIN * NPIX + (size_t)lx_row * NPIX + n0 + lx_col;

    const int wt_off = lw_row * WT_STRIDE + lw_col;
    // interleaved: element (k, n) -> (k>>1)*XT_STRIDE + 2n + (k&1)
    const int xt_off0 = (lx_row >> 1) * XT_STRIDE + lx_col * 2 + (lx_row & 1);
    const int xt_off1 = ((lx_row + 8) >> 1) * XT_STRIDE + lx_col * 2 + ((lx_row + 8) & 1);

    // ---- preload chunk 0 into buffer 0 ----
    {
        float4 wv  = *(const float4*)(wg);
        float4 xv0 = *(const float4*)(xg);
        float4 xv1 = *(const float4*)(xg + (size_t)8 * NPIX);
        float* wdst = &Wt[0][wt_off];
        wdst[0] = wv.x; wdst[1] = wv.y; wdst[2] = wv.z; wdst[3] = wv.w;
        float* x0 = &Xt[0][xt_off0];
        x0[0] = sanitize(xv0.x); x0[2] = sanitize(xv0.y);
        x0[4] = sanitize(xv0.z); x0[6] = sanitize(xv0.w);
        float* x1 = &Xt[0][xt_off1];
        x1[0] = sanitize(xv1.x); x1[2] = sanitize(xv1.y);
        x1[4] = sanitize(xv1.z); x1[6] = sanitize(xv1.w);
    }
    __syncthreads();

    for (int k0 = 0; k0 < CIN; k0 += BK) {
        const int cur = (k0 >> 4) & 1;
        const int nxt = cur ^ 1;
        const bool has_next = (k0 + BK) < CIN;   // uniform (scalar) condition

        // prefetch next chunk into registers (overlaps with WMMA compute)
        float4 wnv, xnv0, xnv1;
        if (has_next) {
            wnv  = *(const float4*)(wg + (k0 + BK));
            xnv0 = *(const float4*)(xg + (size_t)(k0 + BK) * NPIX);
            xnv1 = *(const float4*)(xg + (size_t)(k0 + BK + 8) * NPIX);
        }

        // ---- compute from current buffer ----
        #pragma unroll
        for (int kk = 0; kk < BK; kk += 4) {
            // A fragment: W[wm + l16][kh..kh+1], kh = kk + 2*half -> single b64
            v2f a = *(const v2f*)&Wt[cur][(wm + l16) * WT_STRIDE + kk + 2 * half];

            // B fragments: interleaved pair row P = kk/2 + half, contiguous b64
            const float* xrow = &Xt[cur][((kk >> 1) + half) * XT_STRIDE + (wn + l16) * 2];
            v2f b0 = *(const v2f*)(xrow + 0);
            v2f b1 = *(const v2f*)(xrow + 32);
            v2f b2 = *(const v2f*)(xrow + 64);
            v2f b3 = *(const v2f*)(xrow + 96);

            c0 = __builtin_amdgcn_wmma_f32_16x16x4_f32(false, a, false, b0, (short)0, c0, false, false);
            c1 = __builtin_amdgcn_wmma_f32_16x16x4_f32(false, a, false, b1, (short)0, c1, false, false);
            c2 = __builtin_amdgcn_wmma_f32_16x16x4_f32(false, a, false, b2, (short)0, c2, false, false);
            c3 = __builtin_amdgcn_wmma_f32_16x16x4_f32(false, a, false, b3, (short)0, c3, false, false);
        }

        // ---- stage next chunk into the other buffer ----
        if (has_next) {
            float* wdst = &Wt[nxt][wt_off];
            wdst[0] = wnv.x; wdst[1] = wnv.y; wdst[2] = wnv.z; wdst[3] = wnv.w;
            float* x0 = &Xt[nxt][xt_off0];
            x0[0] = sanitize(xnv0.x); x0[2] = sanitize(xnv0.y);
            x0[4] = sanitize(xnv0.z); x0[6] = sanitize(xnv0.w);
            float* x1 = &Xt[nxt][xt_off1];
            x1[0] = sanitize(xnv1.x); x1[2] = sanitize(xnv1.y);
            x1[4] = sanitize(xnv1.z); x1[6] = sanitize(xnv1.w);
        }
        __syncthreads();
    }

    // ---- epilogue: C/D layout -> VGPR v holds M = v (lanes 0-15) / v+8 (lanes 16-31)
    const int row_base = m0 + wm + (half ? 8 : 0);
    const int col      = n0 + wn + l16;
    float* zb = z + (size_t)b * COUT * NPIX;
    #pragma unroll
    for (int v = 0; v < 8; ++v) {
        float* zr = zb + (size_t)(row_base + v) * NPIX + col;
        zr[0]  = c0[v];
        zr[16] = c1[v];
        zr[32] = c2[v];
        zr[48] = c3[v];
    }
}

// -------------------------------------------------------------------------
// Kernel 2: separable FIR 2x upsample ([1,3,3,1]/4 per axis) + bias
//   out[b,o,2i+di,2j+dj] from 2x2 neighborhood of Z, weights {1,3,9}/16
// Grid: (W/16, H/16, B*COUT), Block: (16,16)
// -------------------------------------------------------------------------
__global__ __launch_bounds__(256) void fir_up2_kernel(
    const float* __restrict__ z,     // [B*COUT, HH, WW]
    const float* __restrict__ bias,  // [COUT]
    float* __restrict__ out)         // [B*COUT, H2, W2]
{
    __shared__ float T[18][19];      // 16x16 tile + 1 halo, padded row

    const int bo = blockIdx.z;
    const int o  = bo & (COUT - 1);
    const int i0 = blockIdx.y * 16;
    const int j0 = blockIdx.x * 16;

    const float* zp = z + (size_t)bo * NPIX;
    const int tid = threadIdx.y * 16 + threadIdx.x;

    for (int t = tid; t < 18 * 18; t += 256) {
        const int li = t / 18, lj = t % 18;
        const int gi = i0 + li - 1, gj = j0 + lj - 1;
        float v = 0.0f;
        if (gi >= 0 && gi < HH && gj >= 0 && gj < WW) v = zp[gi * WW + gj];
        T[li][lj] = v;
    }
    __syncthreads();

    const int ty = threadIdx.y, tx = threadIdx.x;
    const int ci = ty + 1, cj = tx + 1;

    const float nw = T[ci - 1][cj - 1], nn = T[ci - 1][cj], ne = T[ci - 1][cj + 1];
    const float wwv = T[ci][cj - 1],    ee = T[ci][cj],     ev = T[ci][cj + 1];
    const float sw = T[ci + 1][cj - 1], ss = T[ci + 1][cj], se = T[ci + 1][cj + 1];

    const float bb = bias[o];
    const float s = 1.0f / 16.0f;

    const float o00 = (nw + 3.0f * nn + 3.0f * wwv + 9.0f * ee) * s + bb;
    const float o01 = (3.0f * nn + ne + 9.0f * ee + 3.0f * ev) * s + bb;
    const float o10 = (3.0f * wwv + 9.0f * ee + sw + 3.0f * ss) * s + bb;
    const float o11 = (9.0f * ee + 3.0f * ev + 3.0f * ss + se) * s + bb;

    float* op = out + (size_t)bo * H2 * W2;
    const int oy = 2 * (i0 + ty);
    const int ox = 2 * (j0 + tx);

    float2 r0; r0.x = o00; r0.y = o01;
    float2 r1; r1.x = o10; r1.y = o11;
    *(float2*)&op[(size_t)oy * W2 + ox]       = r0;
    *(float2*)&op[(size_t)(oy + 1) * W2 + ox] = r1;
}

extern "C" void kernel_launch(void* const* d_in, const int* in_sizes, int n_in,
                              void* d_out, int out_size, void* d_ws, size_t ws_size,
                              hipStream_t stream) {
    const float* x    = (const float*)d_in[0];  // (16,512,64,64)
    // d_in[1] = fir_kernel (4x4) -- compile-time constant [1,3,3,1] outer /16, folded in
    const float* wmat = (const float*)d_in[2];  // (512,512,1,1)
    const float* bias = (const float*)d_in[3];  // (512,)
    float* out = (float*)d_out;                 // (16,512,128,128)
    float* z   = (float*)d_ws;                  // (16,512,64,64) scratch, 128 MiB

    const int B = 16;

    dim3 g1(NPIX / BN, COUT / BM, B);           // (32, 8, 16)
    gemm_cmix_kernel<<<g1, dim3(256), 0, stream>>>(x, wmat, z);

    dim3 g2(WW / 16, HH / 16, B * COUT);        // (4, 4, 8192)
    fir_up2_kernel<<<g2, dim3(16, 16), 0, stream>>>(z, bias, out);
}